// DOSAConLoss_65249143161375
// MI455X (gfx1250) — compile-verified
//
#include <hip/hip_runtime.h>
#include <hip/hip_bf16.h>
#include <math.h>

#define N_ 8192
#define D_ 128
#define C_ 80

typedef __attribute__((ext_vector_type(2))) float v2f;
typedef __attribute__((ext_vector_type(8))) float v8f;

__global__ void kInit(float* out) {
    if (threadIdx.x == 0 && blockIdx.x == 0) out[0] = 0.0f;
}

// Row squared-norms of embeddings -> workspace
__global__ void kSq(const float* __restrict__ E, float* __restrict__ sq) {
    int i = blockIdx.x * blockDim.x + threadIdx.x;
    if (i < N_) {
        const float* row = E + (size_t)i * D_;
        float s = 0.0f;
#pragma unroll
        for (int k = 0; k < D_; k += 4) {
            float4 v = *(const float4*)(row + k);
            s += v.x * v.x + v.y * v.y + v.z * v.z + v.w * v.w;
        }
        sq[i] = s;
    }
}

// Fused loss_loc (CIoU-based) + focal classification loss
__global__ void kPoint(const float* __restrict__ pb, const float* __restrict__ pc,
                       const float* __restrict__ tb, const float* __restrict__ tc,
                       const float* __restrict__ dens, float* __restrict__ out) {
    __shared__ float sdata[256];
    int tid = threadIdx.x;
    int i = blockIdx.x * blockDim.x + tid;
    float local = 0.0f;
    if (i < N_) {
        float4 b1 = *(const float4*)(pb + 4 * i);
        float4 b2 = *(const float4*)(tb + 4 * i);
        float x1 = b1.x, y1 = b1.y, w1 = b1.z, h1 = b1.w;
        float x2 = b2.x, y2 = b2.y, w2 = b2.z, h2 = b2.w;
        float l1 = x1 - 0.5f * w1, t1 = y1 - 0.5f * h1, r1 = x1 + 0.5f * w1, d1 = y1 + 0.5f * h1;
        float l2 = x2 - 0.5f * w2, t2 = y2 - 0.5f * h2, r2 = x2 + 0.5f * w2, d2b = y2 + 0.5f * h2;
        float iw = fmaxf(fminf(r1, r2) - fmaxf(l1, l2), 0.0f);
        float ih = fmaxf(fminf(d1, d2b) - fmaxf(t1, t2), 0.0f);
        float inter = iw * ih;
        float uni = w1 * h1 + w2 * h2 - inter + 1e-7f;
        float iou = inter / uni;
        float cw = fmaxf(r1, r2) - fminf(l1, l2);
        float ch = fmaxf(d1, d2b) - fminf(t1, t2);
        float c2 = cw * cw + ch * ch + 1e-7f;
        float rho2 = (x2 - x1) * (x2 - x1) + (y2 - y1) * (y2 - y1);
        float dv = atanf(w2 / (h2 + 1e-7f)) - atanf(w1 / (h1 + 1e-7f));
        float v = (4.0f / (3.14159265358979f * 3.14159265358979f)) * dv * dv;
        float alpha_ = v / (v - iou + 1.0f + 1e-7f);
        float ciou = iou - (rho2 / c2 + v * alpha_);
        float saf = powf(1.0f - ciou, 2.5f) / (w2 * h2 + 1e-7f);
        float dw = 1.0f + 1.2f * dens[i];
        float hw = 1.0f / (1.0f + expf(-5.0f * (0.5f - ciou)));
        local += dw * hw * saf * (1.0f / (float)N_);

        // focal cls: mean over C then sum over rows == sum(term)/C
        float fsum = 0.0f;
        const float* prow = pc + (size_t)i * C_;
        const float* trow = tc + (size_t)i * C_;
        for (int c = 0; c < C_; ++c) {
            float x = prow[c], t = trow[c];
            float lse = log1pf(expf(-fabsf(x)));
            float ls_pos = fminf(x, 0.0f) - lse;   // log sigmoid(x)
            float ls_neg = fminf(-x, 0.0f) - lse;  // log sigmoid(-x)
            float bce = -(t * ls_pos + (1.0f - t) * ls_neg);
            float p = 1.0f / (1.0f + expf(-x));
            float pt = t * p + (1.0f - t) * (1.0f - p);
            float mod = powf(1.0f - pt, 1.5f);
            float af = t * 0.25f + (1.0f - t) * 0.75f;
            fsum += bce * mod * af;
        }
        local += fsum * (1.0f / (float)C_);
    }
    sdata[tid] = local;
    __syncthreads();
    for (int s = 128; s > 0; s >>= 1) {
        if (tid < s) sdata[tid] += sdata[tid + s];
        __syncthreads();
    }
    if (tid == 0) atomicAdd(out, sdata[0]);
}

__device__ __forceinline__ float pair_iou(float4 bi, float4 bj) {
    float li = bi.x - 0.5f * bi.z, ti = bi.y - 0.5f * bi.w;
    float ri = bi.x + 0.5f * bi.z, di = bi.y + 0.5f * bi.w;
    float lj = bj.x - 0.5f * bj.z, tj = bj.y - 0.5f * bj.w;
    float rj = bj.x + 0.5f * bj.z, dj = bj.y + 0.5f * bj.w;
    float iw = fmaxf(fminf(ri, rj) - fmaxf(li, lj), 0.0f);
    float ih = fmaxf(fminf(di, dj) - fmaxf(ti, tj), 0.0f);
    float inter = iw * ih;
    float uni = bi.z * bi.w + bj.z * bj.w - inter + 1e-7f;
    return inter / uni;
}

// Contrastive term: Gram tiles via V_WMMA_F32_16X16X4_F32.
// Register blocking: each wave computes a 32x32 output tile (2x2 accumulators)
// -> 4 WMMAs per 4 float2 loads (2x the FLOP/byte of a single-tile wave).
// Block = 256 threads = 8 waves (4 in i, 2 in j) -> 128 x 64 region per block.
__global__ void kPair(const float* __restrict__ E, const float* __restrict__ boxes,
                      const float* __restrict__ sq, float* __restrict__ out) {
    __shared__ float sdata[256];
    int tid = threadIdx.x;
    int wave = tid >> 5;
    int lane = tid & 31;
    int half = lane >> 4;   // K pair selector for A/B fragments, row-half for C/D
    int l16 = lane & 15;
    int wx = wave & 1;      // j direction (2 waves)
    int wy = wave >> 1;     // i direction (4 waves)

    int j0 = blockIdx.x * 64 + wx * 32;
    int i0 = blockIdx.y * 128 + wy * 32;

    float local = 0.0f;
    // Strict upper triangle: tile [i0,i0+32) x [j0,j0+32) contributes iff i0 <= j0
    // (both multiples of 32). Condition is wave-uniform -> EXEC all-1s for WMMA.
    if (i0 <= j0) {
        v8f zero = {0.f, 0.f, 0.f, 0.f, 0.f, 0.f, 0.f, 0.f};
        v8f acc[2][2];
        acc[0][0] = zero; acc[0][1] = zero; acc[1][0] = zero; acc[1][1] = zero;

        // A fragment: 16x4 fp32, lane<16 holds K=0,1 ; lane>=16 holds K=2,3
        const float* a0 = E + (size_t)(i0 + l16) * D_ + 2 * half;
        const float* a1 = a0 + 16 * D_;
        const float* b0 = E + (size_t)(j0 + l16) * D_ + 2 * half;
        const float* b1 = b0 + 16 * D_;
#pragma unroll
        for (int k = 0; k < D_; k += 4) {
            v2f va0 = *(const v2f*)(a0 + k);
            v2f va1 = *(const v2f*)(a1 + k);
            v2f vb0 = *(const v2f*)(b0 + k);
            v2f vb1 = *(const v2f*)(b1 + k);
            acc[0][0] = __builtin_amdgcn_wmma_f32_16x16x4_f32(
                false, va0, false, vb0, (short)0, acc[0][0], false, false);
            acc[0][1] = __builtin_amdgcn_wmma_f32_16x16x4_f32(
                false, va0, false, vb1, (short)0, acc[0][1], false, false);
            acc[1][0] = __builtin_amdgcn_wmma_f32_16x16x4_f32(
                false, va1, false, vb0, (short)0, acc[1][0], false, false);
            acc[1][1] = __builtin_amdgcn_wmma_f32_16x16x4_f32(
                false, va1, false, vb1, (short)0, acc[1][1], false, false);
        }

        // Whole tile strictly above diagonal? Then i<j always holds.
        bool interior = (i0 + 32 <= j0);

        // C/D layout: VGPR r -> i = i0 + 16*ti + r + 8*half ; lane%16 -> j = j0 + 16*tj + l16
#pragma unroll
        for (int tj = 0; tj < 2; ++tj) {
            int j = j0 + 16 * tj + l16;
            float4 bj = *(const float4*)(boxes + 4 * j);
            float sqj = sq[j];
#pragma unroll
            for (int ti = 0; ti < 2; ++ti) {
#pragma unroll
                for (int r = 0; r < 8; ++r) {
                    int i = i0 + 16 * ti + 8 * half + r;
                    if (interior || i < j) {
                        float4 bi = *(const float4*)(boxes + 4 * i);
                        float piou = pair_iou(bi, bj);
                        if (piou > 0.3f) {
                            float dot = acc[ti][tj][r];
                            float d2 = sq[i] + sqj - 2.0f * dot;
                            float dist = sqrtf(fmaxf(d2, 0.0f) + 1e-12f);
                            float h = fmaxf(1.0f - dist, 0.0f);
                            local += h * h;
                        }
                    }
                }
            }
        }
    }
    sdata[tid] = local;
    __syncthreads();
    for (int s = 128; s > 0; s >>= 1) {
        if (tid < s) sdata[tid] += sdata[tid + s];
        __syncthreads();
    }
    if (tid == 0 && sdata[0] != 0.0f) atomicAdd(out, 0.5f * sdata[0]);  // LAMBDA_CONTRAST
}

extern "C" void kernel_launch(void* const* d_in, const int* in_sizes, int n_in,
                              void* d_out, int out_size, void* d_ws, size_t ws_size,
                              hipStream_t stream) {
    const float* pred_boxes   = (const float*)d_in[0];
    const float* pred_cls     = (const float*)d_in[1];
    const float* target_boxes = (const float*)d_in[2];
    const float* target_cls   = (const float*)d_in[3];
    const float* embeddings   = (const float*)d_in[4];
    const float* density_map  = (const float*)d_in[5];
    float* out = (float*)d_out;
    float* sq  = (float*)d_ws;  // N_ floats = 32 KB

    kInit<<<1, 64, 0, stream>>>(out);
    kSq<<<N_ / 256, 256, 0, stream>>>(embeddings, sq);
    kPoint<<<N_ / 256, 256, 0, stream>>>(pred_boxes, pred_cls, target_boxes,
                                         target_cls, density_map, out);
    dim3 gridPair(N_ / 64, N_ / 128);  // 128 x 64 blocks, 8 waves/block, 32x32 tile/wave
    kPair<<<gridPair, 256, 0, stream>>>(embeddings, pred_boxes, sq, out);
}